// Proposal_Sampling_11759620456745
// MI455X (gfx1250) — compile-verified
//
#include <hip/hip_runtime.h>
#include <stdint.h>

// Problem shape (fixed by setup_inputs): B=32, T=64, D=512, K=80.
namespace {
constexpr int BATCH   = 32;
constexpr int T       = 64;
constexpr int TT      = T * T;          // 4096 cells per batch
constexpr int D       = 512;
constexpr int K       = 80;             // MAX_NUM
constexpr int THREADS = 256;            // 8 wave32s per block
constexpr int EPT     = TT / THREADS;   // 16 elements per thread
constexpr int WAVES   = THREADS / 32;   // 8

// Output layout (floats), tuple concatenated flat in return order:
// prop_lists [B,K,D], pred_s_e [B,K,2], offset_gt_list [B,K,2], pred_score [B,K]
constexpr size_t OUT_PROP  = 0;
constexpr size_t OUT_PSE   = (size_t)BATCH * K * D;            // 1310720
constexpr size_t OUT_OFF   = OUT_PSE + (size_t)BATCH * K * 2;  // 1315840
constexpr size_t OUT_SCORE = OUT_OFF + (size_t)BATCH * K * 2;  // 1320960
} // namespace

// 64-bit xor-shuffle max across a wave32 (two 32-bit lane shuffles).
__device__ __forceinline__ uint64_t shfl_xor_max_u64(uint64_t v, int mask) {
  uint32_t lo = (uint32_t)v;
  uint32_t hi = (uint32_t)(v >> 32);
  uint32_t olo = __shfl_xor((unsigned int)lo, mask, 32);
  uint32_t ohi = __shfl_xor((unsigned int)hi, mask, 32);
  uint64_t o = ((uint64_t)ohi << 32) | (uint64_t)olo;
  return (o > v) ? o : v;
}

// Map float bits to a monotonically ordered u32 (larger value -> larger key).
// Exact zeros (the reference masks them to -inf) map to 0 (below everything).
__device__ __forceinline__ uint32_t order_key(float f) {
  uint32_t b = __float_as_uint(f);
  if ((b << 1) == 0u) return 0u;  // +0 / -0 -> masked (acts as -inf)
  return b ^ ((b >> 31) ? 0xFFFFFFFFu : 0x80000000u);
}

// One block per batch: exact descending top-K with lowest-index tie-break
// (matches jax.lax.top_k). Also writes the small outputs and prefetches the
// winning map2d rows into L2 for the gather kernel.
__global__ __launch_bounds__(THREADS)
void topk_select_kernel(const float* __restrict__ logit,      // [B,T,T]
                        const float* __restrict__ offset_gt,  // [B,T,T,2]
                        const float* __restrict__ tmap,       // [B,T,T]
                        const float* __restrict__ map2d,      // [B,T,T,D]
                        float* __restrict__ out,
                        int* __restrict__ ws_idx)              // [B,K]
{
  const int b    = blockIdx.x;
  const int t    = threadIdx.x;
  const int lane = t & 31;
  const int wave = t >> 5;

  // Double-buffered wave partials: two barriers per selection round.
  __shared__ alignas(16) uint64_t s_partial[2][WAVES];
  __shared__ int s_topk[K];

  // Coalesced load: element e of thread t is flat index e*THREADS + t.
  const float* lg = logit + (size_t)b * TT;
  uint64_t key[EPT];
#pragma unroll
  for (int e = 0; e < EPT; ++e) {
    const int idx = e * THREADS + t;
    const uint32_t ord = order_key(lg[idx]);
    // High word: ordered value; low word: (TT-1-idx) so equal values prefer
    // the smaller index (larger packed key). Tombstone = 0 < any live key.
    key[e] = ((uint64_t)ord << 32) | (uint32_t)(TT - 1 - idx);
  }

  // Cached per-thread max; only the tombstoning thread rescans each round.
  uint64_t localmax = key[0];
#pragma unroll
  for (int e = 1; e < EPT; ++e) localmax = (key[e] > localmax) ? key[e] : localmax;

  for (int k = 0; k < K; ++k) {
    // Wave32 reduction of per-thread maxima.
    uint64_t m = localmax;
#pragma unroll
    for (int off = 16; off > 0; off >>= 1) m = shfl_xor_max_u64(m, off);
    if (lane == 0) s_partial[k & 1][wave] = m;
    __syncthreads();

    // Every thread folds the 8 wave partials (broadcast LDS reads).
    uint64_t w = s_partial[k & 1][0];
#pragma unroll
    for (int i = 1; i < WAVES; ++i) {
      const uint64_t p = s_partial[k & 1][i];
      w = (p > w) ? p : w;
    }

    const int widx = (TT - 1) - (int)(uint32_t)(w & 0xFFFFFFFFu);
    if (t == 0) s_topk[k] = widx;

    // Tombstone the winner in its owner's registers (no dynamic reg indexing).
    if ((widx & (THREADS - 1)) == t) {
      const int ehit = widx >> 8;  // widx / THREADS
#pragma unroll
      for (int e = 0; e < EPT; ++e)
        if (e == ehit) key[e] = 0ull;
      uint64_t mm = key[0];
#pragma unroll
      for (int e = 1; e < EPT; ++e) mm = (key[e] > mm) ? key[e] : mm;
      localmax = mm;
    }

    // Warm L2 with the winner's 2KB map2d row for the gather kernel
    // (gfx1250 global_prefetch_b8 path).
    if (t < 16) {
      const char* row =
          (const char*)(map2d + ((size_t)b * TT + (size_t)widx) * D) + (size_t)t * 128;
      __builtin_prefetch(row, 0, 1);
    }

    __syncthreads();  // second barrier per round (partials double-buffered)
  }
  __syncthreads();  // s_topk[K-1] was written before the last round's end barrier

  // Small outputs + winner indices for the gather kernel.
  if (t < K) {
    const int idx = s_topk[t];
    const int row = idx >> 6;  // / T
    const int col = idx & (T - 1);
    const size_t bk = (size_t)b * K + (size_t)t;
    out[OUT_PSE + bk * 2 + 0] = (float)row;
    out[OUT_PSE + bk * 2 + 1] = (float)(col + 1);
    const size_t cell = (size_t)b * TT + (size_t)idx;
    out[OUT_OFF + bk * 2 + 0] = offset_gt[cell * 2 + 0];
    out[OUT_OFF + bk * 2 + 1] = offset_gt[cell * 2 + 1];
    out[OUT_SCORE + bk]       = tmap[cell];
    ws_idx[b * K + t] = idx;
  }
}

// One block per (proposal, batch): 2KB row copy. On gfx1250 with the async
// builtins available, route the copy through the CDNA5 async global<->LDS
// data mover (ASYNCcnt path, no VGPR data movement); otherwise plain float4.
#if defined(__gfx1250__) &&                                         \
    __has_builtin(__builtin_amdgcn_global_load_async_to_lds_b128) && \
    __has_builtin(__builtin_amdgcn_global_store_async_from_lds_b128) && \
    __has_builtin(__builtin_amdgcn_s_wait_asynccnt)
#define USE_ASYNC_LDS_COPY 1
#else
#define USE_ASYNC_LDS_COPY 0
#endif

__global__ __launch_bounds__(128)
void gather_rows_kernel(const float* __restrict__ map2d,  // [B,T,T,D]
                        const int* __restrict__ ws_idx,   // [B,K]
                        float* __restrict__ out)
{
  const int p = blockIdx.x;  // 0..K-1
  const int b = blockIdx.y;  // 0..B-1
  const int idx = ws_idx[b * K + p];
  const float* src =
      map2d + ((size_t)b * TT + (size_t)idx) * D + (size_t)threadIdx.x * 4;
  float* dst =
      out + OUT_PROP + ((size_t)b * K + (size_t)p) * D + (size_t)threadIdx.x * 4;

#if USE_ASYNC_LDS_COPY
  __shared__ alignas(16) float smem[D];  // 2KB staging tile
  // Builtin signature (from clang diagnostics): param 1 is
  // 'int __attribute__((vector_size(16))) __device__ *' i.e. v4i addrspace(1)*.
  typedef int v4i_t __attribute__((vector_size(16)));
  typedef __attribute__((address_space(1))) v4i_t g_v4i;
  typedef __attribute__((address_space(3))) v4i_t l_v4i;
  // Flat == global address for global memory; low 32 bits of a flat LDS
  // address are the wave-relative DS offset (ISA aperture rules).
  g_v4i* g_src = (g_v4i*)(uintptr_t)src;
  g_v4i* g_dst = (g_v4i*)(uintptr_t)dst;
  l_v4i* l_tile =
      (l_v4i*)(uintptr_t)(uint32_t)(uintptr_t)&smem[threadIdx.x * 4];
  __builtin_amdgcn_global_load_async_to_lds_b128(g_src, l_tile, 0, 0);
  __builtin_amdgcn_s_wait_asynccnt(0);  // LDS tile populated
  __builtin_amdgcn_global_store_async_from_lds_b128(g_dst, l_tile, 0, 0);
  __builtin_amdgcn_s_wait_asynccnt(0);  // store data pulled from LDS
#else
  *(float4*)dst = *(const float4*)src;  // 128 threads * 16B = 2KB row
#endif
}

extern "C" void kernel_launch(void* const* d_in, const int* in_sizes, int n_in,
                              void* d_out, int out_size, void* d_ws, size_t ws_size,
                              hipStream_t stream) {
  (void)in_sizes; (void)n_in; (void)out_size; (void)ws_size;
  const float* selection_logit = (const float*)d_in[0];  // [32,64,64]
  const float* map2d           = (const float*)d_in[1];  // [32,64,64,512]
  const float* offset_gt       = (const float*)d_in[2];  // [32,64,64,2]
  const float* tmap            = (const float*)d_in[3];  // [32,64,64]
  float* out   = (float*)d_out;
  int* ws_idx  = (int*)d_ws;  // 32*80 ints of scratch

  topk_select_kernel<<<dim3(BATCH), dim3(THREADS), 0, stream>>>(
      selection_logit, offset_gt, tmap, map2d, out, ws_idx);
  gather_rows_kernel<<<dim3(K, BATCH), dim3(128), 0, stream>>>(
      map2d, ws_idx, out);
}